// DeepEdgeFeatureGAT_44220983280249
// MI455X (gfx1250) — compile-verified
//
#include <hip/hip_runtime.h>
#include <math.h>

// ---------------------------------------------------------------------------
// DeepEdgeFeatureGAT for MI455X (gfx1250, wave32).
// Dense 128x128 GEMMs use V_WMMA_F32_16X16X4_F32 with the A tile staged in
// LDS (bank-conflict-free, 8x global-read reuse). Edge phase is bandwidth-
// optimized: logits via precomputed per-node scalars ss/sd + 16-wide edge
// dot; scatter uses HW global_atomic_add_f32.
// ---------------------------------------------------------------------------

typedef float v2f __attribute__((ext_vector_type(2)));
typedef float v8f __attribute__((ext_vector_type(8)));

#define ALPHA_RES 0.2f
#define XT_PITCH 132  // 128 + 4 pad -> half-wave lanes hit banks 0,4,...,60

// ---------------- ordered-key helpers for float atomicMax --------------------
__device__ __forceinline__ unsigned float_ordkey(float f) {
  unsigned u = __float_as_uint(f);
  return (u & 0x80000000u) ? ~u : (u | 0x80000000u);
}
__device__ __forceinline__ float ordkey_float(unsigned k) {
  unsigned u = (k & 0x80000000u) ? (k ^ 0x80000000u) : ~k;
  return __uint_as_float(u);
}

// ---------------- per-layer small precompute ---------------------------------
// vsrc = Wsrc @ att_lo (128), vdst = Wdst @ att_hi (128), w16 = Wedge @ att_lo
// Also resets the global-max key.
__global__ void precompute_kernel(const float* __restrict__ Wsrc,
                                  const float* __restrict__ Wdst,
                                  const float* __restrict__ Wedge,
                                  const float* __restrict__ att,
                                  float* __restrict__ vsrc,
                                  float* __restrict__ vdst,
                                  float* __restrict__ w16,
                                  unsigned* __restrict__ gmax) {
  int t = threadIdx.x;
  if (t == 0) *gmax = 0u;
  if (t < 128) {
    float s = 0.f, d = 0.f;
    for (int j = 0; j < 128; ++j) {
      s += Wsrc[t * 128 + j] * att[j];
      d += Wdst[t * 128 + j] * att[128 + j];
    }
    vsrc[t] = s;
    vdst[t] = d;
  } else if (t < 144) {
    int r = t - 128;
    float s = 0.f;
    for (int j = 0; j < 128; ++j) s += Wedge[r * 128 + j] * att[j];
    w16[r] = s;
  }
}

// ---------------- fp32 WMMA GEMM:  Y(n,128) = X(n,128) @ W(128,128) ----------
// Block = 256 threads = 8 waves; one wave per 16x16 output tile (8 col tiles).
// The 16x128 A tile is staged once in LDS and shared by all 8 waves.
// A frag (16x4 f32): lanes 0-15 -> M, K={k,k+1}; lanes 16-31 -> M, K={k+2,k+3}
// B frag (4x16 f32): mirrored; C/D: VGPR r -> M = r (+8 for hi lanes), N=lane.
__global__ void gemm128_wmma(const float* __restrict__ X,
                             const float* __restrict__ W,
                             float* __restrict__ Y, int nrows) {
  __shared__ float xt[16 * XT_PITCH];

  const int tid  = threadIdx.x;
  const int wave = tid >> 5;
  const int lane = tid & 31;
  const int m0   = blockIdx.x * 16;

  // cooperative, coalesced load of the 16x128 X tile into LDS
  {
    int r  = tid >> 4;          // 0..15
    int c0 = (tid & 15) * 8;    // 0,8,...,120
    int row = m0 + r;
    if (row >= nrows) row = nrows - 1;
    const float* srcp = X + (size_t)row * 128 + c0;
#pragma unroll
    for (int j = 0; j < 8; ++j) xt[r * XT_PITCH + c0 + j] = srcp[j];
  }
  __syncthreads();

  const int n0    = wave * 16;
  const int mlane = lane & 15;
  const int kh    = (lane >> 4) * 2;
  const float* xl = xt + mlane * XT_PITCH + kh;

  v8f acc = {0.f, 0.f, 0.f, 0.f, 0.f, 0.f, 0.f, 0.f};
#pragma unroll 8
  for (int k = 0; k < 128; k += 4) {
    v2f a;
    a.x = xl[k];
    a.y = xl[k + 1];
    v2f b;
    b.x = W[(size_t)(k + kh) * 128 + n0 + mlane];
    b.y = W[(size_t)(k + kh + 1) * 128 + n0 + mlane];
    acc = __builtin_amdgcn_wmma_f32_16x16x4_f32(false, a, false, b,
                                                (short)0, acc, false, false);
  }

  const int mo = (lane >> 4) * 8;
  float* yr = Y + (size_t)(m0 + mo) * 128 + n0 + mlane;
  if (m0 + 16 <= nrows) {
    // full tile: straight-line stores, immediate offsets
#pragma unroll
    for (int r = 0; r < 8; ++r) yr[(size_t)r * 128] = acc[r];
  } else {
#pragma unroll
    for (int r = 0; r < 8; ++r)
      if (m0 + mo + r < nrows) yr[(size_t)r * 128] = acc[r];
  }
}

// ---------------- per-node attention scalars ss, sd --------------------------
__global__ void node_scores(const float* __restrict__ X,
                            const float* __restrict__ vsrc,
                            const float* __restrict__ vdst,
                            float* __restrict__ ss, float* __restrict__ sd,
                            int N) {
  int wave = threadIdx.x >> 5;
  int lane = threadIdx.x & 31;
  int n = blockIdx.x * (blockDim.x >> 5) + wave;
  if (n >= N) return;
  const float* xr = X + (size_t)n * 128;
  float s = 0.f, d = 0.f;
#pragma unroll
  for (int j = 0; j < 4; ++j) {
    int c = lane + j * 32;
    float xv = xr[c];
    s += xv * vsrc[c];
    d += xv * vdst[c];
  }
#pragma unroll
  for (int off = 16; off > 0; off >>= 1) {
    s += __shfl_down(s, off, 32);
    d += __shfl_down(d, off, 32);
  }
  if (lane == 0) { ss[n] = s; sd[n] = d; }
}

// ---------------- init output with bias, zero denominators -------------------
__global__ void init_out(float* __restrict__ Y, const float* __restrict__ bias,
                         float* __restrict__ denom, int N) {
  long long tid = (long long)blockIdx.x * blockDim.x + threadIdx.x;
  long long tot = (long long)N * 128;
  if (tid < tot) Y[tid] = bias[tid & 127];
  if (tid < N) denom[tid] = 0.f;
}

// ---------------- edge pass 1: logits + global max ---------------------------
__global__ void edge_alpha(const int* __restrict__ src, const int* __restrict__ dst,
                           const float* __restrict__ eattr,
                           const float* __restrict__ ss, const float* __restrict__ sd,
                           const float* __restrict__ w16,
                           float* __restrict__ alpha, unsigned* __restrict__ gmax,
                           int E) {
  __shared__ float sw[16];
  if (threadIdx.x < 16) sw[threadIdx.x] = w16[threadIdx.x];
  __syncthreads();
  int e = blockIdx.x * blockDim.x + threadIdx.x;
  float a = -3.4e38f;
  if (e < E) {
    float v = ss[src[e]] + sd[dst[e]];
    const float* ea = eattr + (size_t)e * 16;
#pragma unroll
    for (int j = 0; j < 16; ++j) v += ea[j] * sw[j];
    v = (v > 0.f) ? v : 0.2f * v;   // leaky_relu(0.2)
    alpha[e] = v;
    a = v;
  }
  unsigned key = float_ordkey(a);
#pragma unroll
  for (int off = 16; off > 0; off >>= 1) {
    unsigned o = __shfl_xor(key, off, 32);
    key = (key > o) ? key : o;
  }
  if ((threadIdx.x & 31) == 0) atomicMax(gmax, key);
}

// ---------------- edge pass 2: exp + segment-sum denominator -----------------
__global__ void edge_exp(const int* __restrict__ dst, float* __restrict__ alpha,
                         const unsigned* __restrict__ gmax,
                         float* __restrict__ denom, int E) {
  int e = blockIdx.x * blockDim.x + threadIdx.x;
  if (e >= E) return;
  float m = ordkey_float(*gmax);
  float ae = __expf(alpha[e] - m);
  alpha[e] = ae;  // in-place: now holds a_exp
  __hip_atomic_fetch_add(&denom[dst[e]], ae, __ATOMIC_RELAXED,
                         __HIP_MEMORY_SCOPE_AGENT);
}

// ---------------- edge pass 3: normalized message scatter --------------------
// One wave per edge; lane 0 fetches the per-edge scalars, broadcast via shfl;
// 4 features per lane, coalesced HW float atomics.
__global__ void edge_scatter(const int* __restrict__ src, const int* __restrict__ dst,
                             const float* __restrict__ aexp,
                             const float* __restrict__ denom,
                             const float* __restrict__ xs,
                             float* __restrict__ Y, int E) {
  int wave = threadIdx.x >> 5;
  int lane = threadIdx.x & 31;
  int e = blockIdx.x * (blockDim.x >> 5) + wave;
  if (e >= E) return;
  int s = 0, d = 0;
  float an = 0.f;
  if (lane == 0) {
    s = src[e];
    d = dst[e];
    an = aexp[e] / (denom[d] + 1e-12f);
  }
  s  = __shfl(s, 0, 32);
  d  = __shfl(d, 0, 32);
  an = __shfl(an, 0, 32);
  const float* xr = xs + (size_t)s * 128;
  float* yr = Y + (size_t)d * 128;
#pragma unroll
  for (int j = 0; j < 4; ++j) {
    int c = lane + j * 32;
    __hip_atomic_fetch_add(&yr[c], an * xr[c], __ATOMIC_RELAXED,
                           __HIP_MEMORY_SCOPE_AGENT);
  }
}

// ---------------- GCNII-style mid-layer combines -----------------------------
__global__ void combine_residual(float* __restrict__ g, const float* __restrict__ x0,
                                 long long n) {
  long long i = (long long)blockIdx.x * blockDim.x + threadIdx.x;
  if (i < n) g[i] = (1.0f - ALPHA_RES) * g[i] + ALPHA_RES * x0[i];
}

__global__ void combine_identity(const float* __restrict__ g,
                                 const float* __restrict__ p,
                                 float* __restrict__ h, float beta, long long n) {
  long long i = (long long)blockIdx.x * blockDim.x + threadIdx.x;
  if (i < n) {
    float v = (1.0f - beta) * g[i] + beta * p[i];
    h[i] = (v > 0.f) ? v : 0.01f * v;  // leaky_relu default slope
  }
}

// ---------------------------------------------------------------------------
// Host-side orchestration
// ---------------------------------------------------------------------------
namespace {

struct Scratch {
  float* xs; float* x0; float* h; float* conv; float* p;
  float* alpha; float* denom; float* ss; float* sd;
  float* vsrc; float* vdst; float* w16; unsigned* gmax;
};

void run_gat_layer(const float* Xin, float* Y,
                   const float* Wsrc_l, const float* Wdst_l, const float* att_l,
                   const float* Wedge_l, const float* bias_l,
                   const int* src, const int* dst, const float* eattr,
                   int N, int E, const Scratch& sc, hipStream_t stream) {
  hipLaunchKernelGGL(precompute_kernel, dim3(1), dim3(256), 0, stream,
                     Wsrc_l, Wdst_l, Wedge_l, att_l,
                     sc.vsrc, sc.vdst, sc.w16, sc.gmax);
  hipLaunchKernelGGL(gemm128_wmma, dim3((N + 15) / 16), dim3(256), 0, stream,
                     Xin, Wsrc_l, sc.xs, N);
  hipLaunchKernelGGL(node_scores, dim3((N + 3) / 4), dim3(128), 0, stream,
                     Xin, sc.vsrc, sc.vdst, sc.ss, sc.sd, N);
  long long tot = (long long)N * 128;
  hipLaunchKernelGGL(init_out, dim3((unsigned)((tot + 255) / 256)), dim3(256), 0,
                     stream, Y, bias_l, sc.denom, N);
  hipLaunchKernelGGL(edge_alpha, dim3((E + 255) / 256), dim3(256), 0, stream,
                     src, dst, eattr, sc.ss, sc.sd, sc.w16, sc.alpha, sc.gmax, E);
  hipLaunchKernelGGL(edge_exp, dim3((E + 255) / 256), dim3(256), 0, stream,
                     dst, sc.alpha, sc.gmax, sc.denom, E);
  hipLaunchKernelGGL(edge_scatter, dim3((E + 7) / 8), dim3(256), 0, stream,
                     src, dst, sc.alpha, sc.denom, sc.xs, Y, E);
}

}  // namespace

extern "C" void kernel_launch(void* const* d_in, const int* in_sizes, int n_in,
                              void* d_out, int out_size, void* d_ws, size_t ws_size,
                              hipStream_t stream) {
  const float* x      = (const float*)d_in[0];
  const int*   eidx   = (const int*)d_in[1];
  const float* eattr  = (const float*)d_in[2];
  const float* Wsrc   = (const float*)d_in[3];
  const float* Wdst   = (const float*)d_in[4];
  const float* att    = (const float*)d_in[5];
  const float* Wedge  = (const float*)d_in[6];
  const float* bias   = (const float*)d_in[7];
  const float* midW   = (const float*)d_in[8];

  const int N = in_sizes[0] / 128;
  const int E = in_sizes[1] / 2;
  const int* src = eidx;
  const int* dst = eidx + E;

  // workspace layout
  float* ws = (float*)d_ws;
  size_t NF = (size_t)N * 128;
  Scratch sc;
  sc.xs    = ws;                 // NF
  sc.x0    = sc.xs + NF;         // NF
  sc.h     = sc.x0 + NF;         // NF
  sc.conv  = sc.h + NF;          // NF
  sc.p     = sc.conv + NF;       // NF
  sc.alpha = sc.p + NF;          // E
  sc.denom = sc.alpha + E;       // N
  sc.ss    = sc.denom + N;       // N
  sc.sd    = sc.ss + N;          // N
  sc.vsrc  = sc.sd + N;          // 128
  sc.vdst  = sc.vsrc + 128;      // 128
  sc.w16   = sc.vdst + 128;      // 16
  sc.gmax  = (unsigned*)(sc.w16 + 16);

  const size_t WSZ = 128 * 128;

  // ---- layer 0: x0 = gat_conv(x) ----
  run_gat_layer(x, sc.x0, Wsrc, Wdst, att, Wedge, bias,
                src, dst, eattr, N, E, sc, stream);

  // ---- mid blocks ----
  const float* hcur = sc.x0;
  long long tot = (long long)NF;
  for (int i = 0; i < 3; ++i) {
    int l = i + 1;
    run_gat_layer(hcur, sc.conv,
                  Wsrc + (size_t)l * WSZ, Wdst + (size_t)l * WSZ,
                  att + (size_t)l * 256, Wedge + (size_t)l * 16 * 128,
                  bias + (size_t)l * 128,
                  src, dst, eattr, N, E, sc, stream);
    // g = (1-a)*conv + a*x0   (in place in conv)
    hipLaunchKernelGGL(combine_residual, dim3((unsigned)((tot + 255) / 256)),
                       dim3(256), 0, stream, sc.conv, sc.x0, tot);
    // p = g @ midW[i]
    hipLaunchKernelGGL(gemm128_wmma, dim3((N + 15) / 16), dim3(256), 0, stream,
                       sc.conv, midW + (size_t)i * WSZ, sc.p, N);
    // h = leaky_relu((1-beta)*g + beta*p, 0.01)
    float beta = logf(0.2f / (float)(i + 1) + 1.0f);
    hipLaunchKernelGGL(combine_identity, dim3((unsigned)((tot + 255) / 256)),
                       dim3(256), 0, stream, sc.conv, sc.p, sc.h, beta, tot);
    hcur = sc.h;
  }

  // ---- final layer -> d_out ----
  int l = 4;
  run_gat_layer(hcur, (float*)d_out,
                Wsrc + (size_t)l * WSZ, Wdst + (size_t)l * WSZ,
                att + (size_t)l * 256, Wedge + (size_t)l * 16 * 128,
                bias + (size_t)l * 128,
                src, dst, eattr, N, E, sc, stream);
}